// modifiedPNAnet_80264348827998
// MI455X (gfx1250) — compile-verified
//
#include <hip/hip_runtime.h>
#include <math.h>

typedef float v2f __attribute__((ext_vector_type(2)));
typedef float v8f __attribute__((ext_vector_type(8)));

#define NNODES 50000
#define NEDGES 800000
#define CDIM 64
#define EW_WAVES 4
#define ASTRIDE 196   // floats; 196*4 = 784 B, 16B aligned per row

// ---- async global->LDS (gfx1250) with graceful fallback ----
#define TRY_ASYNC 1
#if TRY_ASYNC && defined(__has_builtin)
#if __has_builtin(__builtin_amdgcn_global_load_async_to_lds_b128)
#define HAVE_ASYNC 1
#endif
#endif
#ifndef HAVE_ASYNC
#define HAVE_ASYNC 0
#endif

// exact pointee type of the builtin's params: 16-byte int vector
typedef int b128v __attribute__((vector_size(16)));
typedef __attribute__((address_space(1))) b128v gb128;  // global (__device__)
typedef __attribute__((address_space(3))) b128v lb128;  // LDS (__shared__)

// ---------- helpers ----------

// order-preserving float <-> uint mapping for atomic min/max
__device__ __forceinline__ unsigned encf(float f) {
  unsigned b = __float_as_uint(f);
  return (b & 0x80000000u) ? ~b : (b | 0x80000000u);
}
__device__ __forceinline__ float decf(unsigned u) {
  unsigned b = (u & 0x80000000u) ? (u & 0x7fffffffu) : ~u;
  return __uint_as_float(b);
}

__device__ __forceinline__ void atomAddF(float* p, float v) {
  __hip_atomic_fetch_add(p, v, __ATOMIC_RELAXED, __HIP_MEMORY_SCOPE_AGENT);
}

// wave-internal LDS producer->consumer sync (cross-lane, same wave)
__device__ __forceinline__ void wave_lds_sync() {
  __builtin_amdgcn_wave_barrier();
  asm volatile("s_wait_dscnt 0x0" ::: "memory");
  __builtin_amdgcn_wave_barrier();
}
// sync that also covers async global->LDS transfers
__device__ __forceinline__ void wave_stage_sync() {
  __builtin_amdgcn_wave_barrier();
#if HAVE_ASYNC
  asm volatile("s_wait_asynccnt 0x0" ::: "memory");
#endif
  asm volatile("s_wait_dscnt 0x0" ::: "memory");
  __builtin_amdgcn_wave_barrier();
}

// fp32 WMMA: D = A(16x4) * B(4x16) + C(16x16)
__device__ __forceinline__ v8f wmma4(v2f a, v2f b, v8f c) {
  return __builtin_amdgcn_wmma_f32_16x16x4_f32(false, a, false, b, (short)0, c,
                                               false, false);
}

// ---------- kernel 0: init workspace ----------
__global__ void k_init(float* __restrict__ s1, float* __restrict__ s2,
                       unsigned* __restrict__ mnk, unsigned* __restrict__ mxk,
                       float* __restrict__ cnt, float* __restrict__ avgsum) {
  size_t i = (size_t)blockIdx.x * blockDim.x + threadIdx.x;
  size_t NC = (size_t)NNODES * CDIM;
  if (i < NC) {
    s1[i] = 0.0f;
    s2[i] = 0.0f;
    mnk[i] = 0xFFFFFFFFu;  // min identity under encf order
    mxk[i] = 0u;           // max identity
  }
  if (i < NNODES) cnt[i] = 0.0f;
  if (i == 0) avgsum[0] = 0.0f;
}

// ---------- kernel P: repack weights into WMMA-B fragment pairs ----------
// pk[(k>>1)*nc + c] = { W[k][c], W[k+1][c] }  -> single b64 per B fragment
__global__ void k_pack(const float* __restrict__ src, v2f* __restrict__ dst,
                       int nkh, int nc) {  // nkh = nk/2
  int i = blockIdx.x * blockDim.x + threadIdx.x;
  int tot = nkh * nc;
  if (i >= tot) return;
  int kp = i / nc, c = i - kp * nc;
  v2f v;
  v.x = src[(size_t)(2 * kp) * nc + c];
  v.y = src[(size_t)(2 * kp + 1) * nc + c];
  dst[i] = v;
}

// ---------- kernel A: avg_log reduction ----------
__global__ void k_avg(const float* __restrict__ deg_hist,
                      float* __restrict__ avgsum) {
  int i = blockIdx.x * blockDim.x + threadIdx.x;
  float v = 0.0f;
  if (i < NNODES) v = logf(deg_hist[i] + 1.0f);
  for (int off = 16; off > 0; off >>= 1) v += __shfl_down(v, off, 32);
  if ((threadIdx.x & 31) == 0) atomAddF(avgsum, v);
}

// ---------- kernel 1: m = x @ W  (16-row tile per wave) ----------
__global__ __launch_bounds__(128) void k_gemm_m(const float* __restrict__ x,
                                                const v2f* __restrict__ pW,
                                                float* __restrict__ m,
                                                int ntiles) {
  int wave = blockIdx.x * (blockDim.x >> 5) + (threadIdx.x >> 5);
  if (wave >= ntiles) return;  // wave-uniform
  int lane = threadIdx.x & 31;
  int n = lane & 15;
  int hi = lane >> 4;
  int koff = hi * 2, hi8 = hi * 8;
  int row = wave * 16 + n;  // A-matrix row M for this lane
  v8f acc[4] = {};
  for (int k0 = 0; k0 < 64; k0 += 4) {
    int k = k0 + koff;
    v2f a = *(const v2f*)(x + (size_t)row * 64 + k);
    int bi = (k >> 1) * 64 + n;
#pragma unroll
    for (int t = 0; t < 4; ++t) acc[t] = wmma4(a, pW[bi + t * 16], acc[t]);
  }
#pragma unroll
  for (int t = 0; t < 4; ++t)
#pragma unroll
    for (int v = 0; v < 8; ++v) {
      int M = v + hi8;
      m[(size_t)(wave * 16 + M) * 64 + t * 16 + n] = acc[t][v];
    }
}

// ---------- kernel 2: edge GEMM + segment reductions ----------
__global__ __launch_bounds__(EW_WAVES * 32) void k_edge(
    const float* __restrict__ m, const float* __restrict__ edge_attr,
    const int* __restrict__ eidx, const float* __restrict__ We,
    const float* __restrict__ be, const v2f* __restrict__ pWpre,
    const float* __restrict__ bpre, float* __restrict__ s1,
    float* __restrict__ s2, unsigned* __restrict__ mnk,
    unsigned* __restrict__ mxk, float* __restrict__ cnt) {
  __shared__ __align__(16) float Ash[EW_WAVES][16 * ASTRIDE];
  __shared__ int dsh[EW_WAVES][16];
  __shared__ float wsh[EW_WAVES][16];
  int wv = threadIdx.x >> 5;
  int tile = blockIdx.x * EW_WAVES + wv;
  int lane = threadIdx.x & 31;
  float* A = Ash[wv];

  // ---- stage RAW A = [m[dst] | m[src] | attr@We+be]; ew applied in epilogue
  int e = lane >> 1, half = lane & 1;
  int ge = tile * 16 + e;
  int srcn = eidx[ge];
  int dstn = eidx[NEDGES + ge];
  float4 ea4 = ((const float4*)edge_attr)[ge];
  float w = ea4.w;
  if (half == 0) {
    dsh[wv][e] = dstn;
    wsh[wv][e] = w;
    atomAddF(&cnt[dstn], 1.0f);
  }
  int c0 = half * 32;
  const float* gi = m + (size_t)dstn * 64 + c0;
  const float* gj = m + (size_t)srcn * 64 + c0;
  float* li = A + e * ASTRIDE + c0;
  float* lj = A + e * ASTRIDE + 64 + c0;
#if HAVE_ASYNC
#pragma unroll
  for (int q = 0; q < 8; ++q)
    __builtin_amdgcn_global_load_async_to_lds_b128(
        (gb128*)(uintptr_t)(gi + q * 4), (lb128*)(uintptr_t)(li + q * 4), 0, 0);
#pragma unroll
  for (int q = 0; q < 8; ++q)
    __builtin_amdgcn_global_load_async_to_lds_b128(
        (gb128*)(uintptr_t)(gj + q * 4), (lb128*)(uintptr_t)(lj + q * 4), 0, 0);
#else
#pragma unroll
  for (int q = 0; q < 8; ++q) ((float4*)li)[q] = ((const float4*)gi)[q];
#pragma unroll
  for (int q = 0; q < 8; ++q) ((float4*)lj)[q] = ((const float4*)gj)[q];
#endif
  {
    float4* Aea = (float4*)(A + e * ASTRIDE + 128 + c0);
#pragma unroll
    for (int q = 0; q < 8; ++q) {
      float4 w0 = ((const float4*)(We + 0 * 64 + c0))[q];
      float4 w1 = ((const float4*)(We + 1 * 64 + c0))[q];
      float4 w2 = ((const float4*)(We + 2 * 64 + c0))[q];
      float4 bb = ((const float4*)(be + c0))[q];
      float4 r;
      r.x = ea4.x * w0.x + ea4.y * w1.x + ea4.z * w2.x + bb.x;
      r.y = ea4.x * w0.y + ea4.y * w1.y + ea4.z * w2.y + bb.y;
      r.z = ea4.x * w0.z + ea4.y * w1.z + ea4.z * w2.z + bb.z;
      r.w = ea4.x * w0.w + ea4.y * w1.w + ea4.z * w2.w + bb.w;
      Aea[q] = r;
    }
  }
  wave_stage_sync();

  // ---- h_raw = A_raw @ Wpre via fp32 WMMA (K = 192) ----
  int n = lane & 15, hi = lane >> 4;
  int koff = hi * 2, hi8 = hi * 8;
  const float* Arow = A + n * ASTRIDE;
  v8f acc[4] = {};
  for (int k0 = 0; k0 < 192; k0 += 4) {
    int k = k0 + koff;
    v2f a = *(const v2f*)(Arow + k);
    int bi = (k >> 1) * 64 + n;
#pragma unroll
    for (int t = 0; t < 4; ++t) acc[t] = wmma4(a, pWpre[bi + t * 16], acc[t]);
  }

  // ---- h = ew * h_raw + bpre; scatter into segment reductions ----
  float wrow[8];
  int drow[8];
#pragma unroll
  for (int v = 0; v < 8; ++v) {
    int M = v + hi8;
    wrow[v] = wsh[wv][M];
    drow[v] = dsh[wv][M];
  }
#pragma unroll
  for (int t = 0; t < 4; ++t) {
    int col = t * 16 + n;
    float bias = bpre[col];
#pragma unroll
    for (int v = 0; v < 8; ++v) {
      float h = wrow[v] * acc[t][v] + bias;
      size_t idx = (size_t)drow[v] * 64 + col;
      atomAddF(&s1[idx], h);
      atomAddF(&s2[idx], h * h);
      atomicMin(&mnk[idx], encf(h));
      atomicMax(&mxk[idx], encf(h));
    }
  }
}

// ---------- kernel 3: post GEMM + GRU (16-node tile per wave/block) ----------
__global__ __launch_bounds__(32) void k_post(
    const float* __restrict__ x, const float* __restrict__ m,
    const float* __restrict__ s1, const float* __restrict__ s2,
    const unsigned* __restrict__ mnk, const unsigned* __restrict__ mxk,
    const float* __restrict__ cnt, const float* __restrict__ avgsum,
    const v2f* __restrict__ pWpost, const float* __restrict__ bpost,
    const v2f* __restrict__ pWih, const float* __restrict__ bih,
    const v2f* __restrict__ pWhh, const float* __restrict__ bhh,
    float* __restrict__ out) {
  __shared__ __align__(16) float outL[16][68];
  __shared__ __align__(16) float rL[16][68];
  __shared__ __align__(16) float zL[16][68];
  __shared__ float invsafeL[16], logdL[16], hasL[16];
  int tile = blockIdx.x;
  int lane = threadIdx.x;
  int n = lane & 15, hi = lane >> 4;
  int koff = hi * 2, hi8 = hi * 8;
  int base = tile * 16;
  float avg_log = avgsum[0] * (1.0f / (float)NNODES);

  if (lane < 16) {
    float c = cnt[base + lane];
    float safe = fmaxf(c, 1.0f);
    invsafeL[lane] = 1.0f / safe;
    logdL[lane] = logf(safe + 1.0f);
    hasL[lane] = (c > 0.0f) ? 1.0f : 0.0f;
  }
  for (int q = lane; q < 256; q += 32) {  // outL = bpost (row-broadcast)
    int row = q >> 4, c4 = q & 15;
    ((float4*)&outL[row][0])[c4] = ((const float4*)bpost)[c4];
  }
  wave_lds_sync();

  int node = base + n;  // A-matrix row (node) for this lane
  float inv_l = invsafeL[n];
  float has_l = hasL[n];

  // ---- out = m@Wp0 + o1@Wp1 + r.*(o1@Wp2) + (1/r).*(o1@Wp3) + bpost ----
  for (int p = 0; p < 3; ++p) {
    v8f acc[4] = {};
    if (p == 0) {  // m @ Wpost[0:64]
      for (int k0 = 0; k0 < 64; k0 += 4) {
        int k = k0 + koff;
        v2f a = *(const v2f*)(m + (size_t)node * 64 + k);
        int bi = (k >> 1) * 64 + n;
#pragma unroll
        for (int t = 0; t < 4; ++t)
          acc[t] = wmma4(a, pWpost[bi + t * 16], acc[t]);
      }
    }
    int rowoff0 = 64 + p * 256;
    for (int s = 0; s < 4; ++s) {  // mean, min, max, std blocks
      int rowoff = rowoff0 + s * 64;
      for (int k0 = 0; k0 < 64; k0 += 4) {
        int k = k0 + koff;
        size_t idx0 = (size_t)node * 64 + k;
        v2f a;
        if (s == 0) {
          v2f s1v = *(const v2f*)(s1 + idx0);
          a.x = s1v.x * inv_l;
          a.y = s1v.y * inv_l;
        } else if (s == 1) {
          uint2 mv = *(const uint2*)(mnk + idx0);
          a.x = (has_l > 0.5f) ? decf(mv.x) : 0.0f;
          a.y = (has_l > 0.5f) ? decf(mv.y) : 0.0f;
        } else if (s == 2) {
          uint2 mv = *(const uint2*)(mxk + idx0);
          a.x = (has_l > 0.5f) ? decf(mv.x) : 0.0f;
          a.y = (has_l > 0.5f) ? decf(mv.y) : 0.0f;
        } else {
          v2f s1v = *(const v2f*)(s1 + idx0);
          v2f s2v = *(const v2f*)(s2 + idx0);
          float m0 = s1v.x * inv_l, m1 = s1v.y * inv_l;
          a.x = sqrtf(fmaxf(s2v.x * inv_l - m0 * m0, 0.0f) + 1e-5f);
          a.y = sqrtf(fmaxf(s2v.y * inv_l - m1 * m1, 0.0f) + 1e-5f);
        }
        int bi = ((rowoff + k) >> 1) * 64 + n;
#pragma unroll
        for (int t = 0; t < 4; ++t)
          acc[t] = wmma4(a, pWpost[bi + t * 16], acc[t]);
      }
    }
#pragma unroll
    for (int t = 0; t < 4; ++t) {
      int col = t * 16 + n;
#pragma unroll
      for (int v = 0; v < 8; ++v) {
        int M = v + hi8;
        float f = 1.0f;
        if (p == 1) f = logdL[M] / avg_log;
        else if (p == 2) f = avg_log / logdL[M];
        outL[M][col] += f * acc[t][v];
      }
    }
  }
  wave_lds_sync();

  // ---- GRU gates r, z ----
  for (int g = 0; g < 2; ++g) {
    for (int t = 0; t < 4; ++t) {
      v8f ai = {}, ah = {};
      for (int k0 = 0; k0 < 64; k0 += 4) {
        int k = k0 + koff;
        v2f a1 = *(const v2f*)(&outL[n][k]);
        v2f a2 = *(const v2f*)(x + (size_t)node * 64 + k);
        int gc = g * 64 + t * 16 + n;
        int bi = (k >> 1) * 192 + gc;
        ai = wmma4(a1, pWih[bi], ai);
        ah = wmma4(a2, pWhh[bi], ah);
      }
      int col = t * 16 + n;
      int gc = g * 64 + col;
      float bsum = bih[gc] + bhh[gc];
#pragma unroll
      for (int v = 0; v < 8; ++v) {
        int M = v + hi8;
        float pre = ai[v] + ah[v] + bsum;
        float gate = 1.0f / (1.0f + expf(-pre));
        if (g == 0) rL[M][col] = gate; else zL[M][col] = gate;
      }
    }
  }
  wave_lds_sync();

  // ---- n gate + output ----
  for (int t = 0; t < 4; ++t) {
    v8f ai = {}, ah = {};
    for (int k0 = 0; k0 < 64; k0 += 4) {
      int k = k0 + koff;
      v2f a1 = *(const v2f*)(&outL[n][k]);
      v2f a2 = *(const v2f*)(x + (size_t)node * 64 + k);
      int gc = 128 + t * 16 + n;
      int bi = (k >> 1) * 192 + gc;
      ai = wmma4(a1, pWih[bi], ai);
      ah = wmma4(a2, pWhh[bi], ah);
    }
    int col = t * 16 + n;
    int gc = 128 + col;
    float bi_ = bih[gc], bh_ = bhh[gc];
#pragma unroll
    for (int v = 0; v < 8; ++v) {
      int M = v + hi8;
      float i_n = ai[v] + bi_;
      float h_n = ah[v] + bh_;
      float r = rL[M][col];
      float z = zL[M][col];
      float nn = tanhf(i_n + r * h_n);
      int nodeM = base + M;
      float xv = x[(size_t)nodeM * 64 + col];
      out[(size_t)nodeM * 64 + col] = (1.0f - z) * nn + z * xv;
    }
  }
}

// ---------- host launcher ----------
extern "C" void kernel_launch(void* const* d_in, const int* in_sizes, int n_in,
                              void* d_out, int out_size, void* d_ws,
                              size_t ws_size, hipStream_t stream) {
  const float* x = (const float*)d_in[0];
  const float* edge_attr = (const float*)d_in[1];
  const float* deg_hist = (const float*)d_in[2];
  const float* W = (const float*)d_in[3];
  const float* We = (const float*)d_in[4];
  const float* be = (const float*)d_in[5];
  const float* Wpre = (const float*)d_in[6];
  const float* bpre = (const float*)d_in[7];
  const float* Wpost = (const float*)d_in[8];
  const float* bpost = (const float*)d_in[9];
  const float* Wih = (const float*)d_in[10];
  const float* bih = (const float*)d_in[11];
  const float* Whh = (const float*)d_in[12];
  const float* bhh = (const float*)d_in[13];
  const int* eidx = (const int*)d_in[14];
  float* out = (float*)d_out;

  size_t NC = (size_t)NNODES * CDIM;
  char* ws = (char*)d_ws;
  float* s1 = (float*)ws;           ws += NC * 4;
  float* s2 = (float*)ws;           ws += NC * 4;
  unsigned* mnk = (unsigned*)ws;    ws += NC * 4;
  unsigned* mxk = (unsigned*)ws;    ws += NC * 4;
  float* cnt = (float*)ws;          ws += ((NNODES + 63) / 64) * 64 * 4;
  float* avgsum = (float*)ws;       ws += 256;
  v2f* pW = (v2f*)ws;               ws += (size_t)32 * 64 * 8;    // 64x64
  v2f* pWpre = (v2f*)ws;            ws += (size_t)96 * 64 * 8;    // 192x64
  v2f* pWpost = (v2f*)ws;           ws += (size_t)416 * 64 * 8;   // 832x64
  v2f* pWih = (v2f*)ws;             ws += (size_t)32 * 192 * 8;   // 64x192
  v2f* pWhh = (v2f*)ws;             ws += (size_t)32 * 192 * 8;   // 64x192
  float* m = out;  // m = x@W lives in d_out; each k_post wave reads only its
                   // own tile's m rows before overwriting them.

  int ntiles_n = NNODES / 16;  // 3125
  k_init<<<(int)((NC + 255) / 256), 256, 0, stream>>>(s1, s2, mnk, mxk, cnt,
                                                      avgsum);
  k_pack<<<(32 * 64 + 255) / 256, 256, 0, stream>>>(W, pW, 32, 64);
  k_pack<<<(96 * 64 + 255) / 256, 256, 0, stream>>>(Wpre, pWpre, 96, 64);
  k_pack<<<(416 * 64 + 255) / 256, 256, 0, stream>>>(Wpost, pWpost, 416, 64);
  k_pack<<<(32 * 192 + 255) / 256, 256, 0, stream>>>(Wih, pWih, 32, 192);
  k_pack<<<(32 * 192 + 255) / 256, 256, 0, stream>>>(Whh, pWhh, 32, 192);
  k_avg<<<(NNODES + 255) / 256, 256, 0, stream>>>(deg_hist, avgsum);
  k_gemm_m<<<(ntiles_n + 3) / 4, 128, 0, stream>>>(x, pW, m, ntiles_n);
  k_edge<<<NEDGES / 16 / EW_WAVES, EW_WAVES * 32, 0, stream>>>(
      m, edge_attr, eidx, We, be, pWpre, bpre, s1, s2, mnk, mxk, cnt);
  k_post<<<ntiles_n, 32, 0, stream>>>(x, m, s1, s2, mnk, mxk, cnt, avgsum,
                                      pWpost, bpost, pWih, bih, pWhh, bhh, out);
}